// Langevin_2439541424651
// MI455X (gfx1250) — compile-verified
//
#include <hip/hip_runtime.h>
#include <cstdint>
#include <cmath>

// Problem constants (match reference)
#define NSAMP   1024
#define DX      128
#define STEPS   200
#define ELEMS   (NSAMP * DX)      // 131072 scan lanes
#define BLOCK   256
#define NSLOTS  4                 // LDS ring slots for noise
#define AHEAD   3                 // async prefetch depth (in-flight ops)

// ---- CDNA5 async global->LDS copy (ASYNCcnt-tracked DMA into LDS) ----------
// GV addressing mode: 64-bit per-lane global address in a VGPR pair, saddr=off.
// VDST operand carries the per-lane LDS byte offset (hardware adds LDS_BASE).
__device__ __forceinline__ void async_load_noise_b32(uint32_t lds_byte_off,
                                                     uint64_t global_addr) {
    asm volatile("global_load_async_to_lds_b32 %0, %1, off"
                 :
                 : "v"(lds_byte_off), "v"(global_addr)
                 : "memory");
}

__device__ __forceinline__ void wait_async_le2() {
    asm volatile("s_wait_asynccnt 0x2" ::: "memory");
}
__device__ __forceinline__ void wait_async_0() {
    asm volatile("s_wait_asynccnt 0x0" ::: "memory");
}

// ---------------------------------------------------------------------------
// Main scan kernel: one thread owns one (n, d) element for all `nsteps`
// steps. x stays in a register; noise streams through a 4-slot LDS ring
// filled by the async DMA engine 3 steps ahead; outputs leave via
// non-temporal stores (315 MB of output > 192 MB L2).
//
// `nsteps` is a RUNTIME argument (host passes STEPS) purely so the compiler
// keeps the loop rolled: the fully-unrolled 200-step body was ~28 KB of
// straight-line code, which busts the I$ for zero pipelining benefit.
// ---------------------------------------------------------------------------
__global__ __launch_bounds__(BLOCK) void langevin_scan(
    const float* __restrict__ x0,
    const float* __restrict__ y0,
    const float* __restrict__ mean_f,
    const float* __restrict__ var_f,
    const float* __restrict__ gammas,
    const float* __restrict__ noise,
    float* __restrict__ x_tot,
    float* __restrict__ y_tot,
    float* __restrict__ out,
    int nsteps)
{
    __shared__ float2 s_gs[STEPS];            // {gamma, sqrt(2*gamma)}
    __shared__ float  s_z[NSLOTS][BLOCK];     // noise ring (wave-private slots)

    const int t = threadIdx.x;

    // Precompute per-step scalars once per block (single pass: BLOCK >= STEPS).
    if (t < nsteps) {
        float g = gammas[t];
        s_gs[t] = make_float2(g, sqrtf(2.0f * g));
    }
    __syncthreads();

    const int gid = blockIdx.x * BLOCK + t;   // element id in [0, N*D)
    const int d   = gid & (DX - 1);
    const int n   = gid >> 7;                 // DX == 128

    const float m  = mean_f[d];
    const float iv = 1.0f / var_f[d];
    float x        = x0[gid];
    const float yv = y0[gid];                 // D_Y == D_X, same flat index

    // LDS byte address of this lane's ring slot 0 (low 32 bits of the
    // generic pointer == workgroup-relative LDS offset).
    const uint32_t lbase = (uint32_t)(uintptr_t)&s_z[0][t];

    // Per-lane global byte address of noise[0][n][d]; step stride is N*D*4.
    uint64_t ga_issue = (uint64_t)(uintptr_t)(noise + gid);
    const uint64_t kstride = (uint64_t)ELEMS * sizeof(float);

    // Prologue: put AHEAD noise steps in flight.
    #pragma unroll
    for (int k = 0; k < AHEAD; ++k) {
        async_load_noise_b32(lbase + (uint32_t)((k & (NSLOTS - 1)) * (BLOCK * 4)),
                             ga_issue);
        ga_issue += kstride;
    }

    // Output base: layout [N, STEPS, D] -> n*STEPS*D + k*D + d
    size_t ob = (size_t)n * (STEPS * DX) + d;
    float* __restrict__ px = x_tot + ob;
    float* __restrict__ py = y_tot + ob;
    float* __restrict__ po = out   + ob;

    // Steady state: branch-free rolled loop; always 3 async ops in flight.
    int k = 0;
    #pragma unroll 2
    for (; k < nsteps - AHEAD; ++k) {
        wait_async_le2();                       // oldest (step k) has landed
        const float z = s_z[k & (NSLOTS - 1)][t];

        // Refill AHEAD steps out. The slot being overwritten was consumed
        // (DS data returned to VGPR) at iteration k-1, before this point.
        async_load_noise_b32(
            lbase + (uint32_t)(((k + AHEAD) & (NSLOTS - 1)) * (BLOCK * 4)),
            ga_issue);
        ga_issue += kstride;

        const float2 gs = s_gs[k];
        const float c  = gs.x * iv;             // gamma / var
        const float sz = gs.y * z;              // sqrt(2*gamma) * z
        // t_old = x - c*(x-m);  x_new = t_old + sz
        const float xn = fmaf(-c, x - m, x) + sz;
        // out = t_old - t_new = c*(x_new - m) - sz
        const float o  = fmaf(c, xn - m, -sz);

        __builtin_nontemporal_store(xn, px);
        __builtin_nontemporal_store(yv, py);
        __builtin_nontemporal_store(o,  po);
        px += DX; py += DX; po += DX;
        x = xn;
    }

    // Drain: last AHEAD steps, no refills.
    for (; k < nsteps; ++k) {
        wait_async_0();
        const float z = s_z[k & (NSLOTS - 1)][t];

        const float2 gs = s_gs[k];
        const float c  = gs.x * iv;
        const float sz = gs.y * z;
        const float xn = fmaf(-c, x - m, x) + sz;
        const float o  = fmaf(c, xn - m, -sz);

        __builtin_nontemporal_store(xn, px);
        __builtin_nontemporal_store(yv, py);
        __builtin_nontemporal_store(o,  po);
        px += DX; py += DX; po += DX;
        x = xn;
    }
}

// steps_expanded: [N, STEPS, 1] int32, value = step index.
// Grid: (ceil(STEPS/BLOCK), N) so the value is just the flat s index.
__global__ __launch_bounds__(BLOCK) void write_steps(int* __restrict__ st,
                                                     int nsteps) {
    const int s = blockIdx.x * BLOCK + threadIdx.x;
    const int n = blockIdx.y;
    if (s < nsteps) {
        __builtin_nontemporal_store(s, st + (size_t)n * nsteps + s);
    }
}

// ---------------------------------------------------------------------------
extern "C" void kernel_launch(void* const* d_in, const int* in_sizes, int n_in,
                              void* d_out, int out_size, void* d_ws, size_t ws_size,
                              hipStream_t stream) {
    (void)in_sizes; (void)n_in; (void)out_size; (void)d_ws; (void)ws_size;

    const float* x0     = (const float*)d_in[0];  // init_samples_x [N, DX]
    const float* y0     = (const float*)d_in[1];  // init_samples_y [N, DY]
    const float* mean_f = (const float*)d_in[2];  // [DX]
    const float* var_f  = (const float*)d_in[3];  // [DX]
    const float* gammas = (const float*)d_in[4];  // [STEPS]
    const float* noise  = (const float*)d_in[5];  // [STEPS, N, DX]

    const size_t big = (size_t)NSAMP * STEPS * DX;   // 26,214,400 floats each
    float* x_tot = (float*)d_out;
    float* y_tot = x_tot + big;
    float* outp  = y_tot + big;
    int*   steps = (int*)(outp + big);               // int32 bits in d_out tail

    langevin_scan<<<ELEMS / BLOCK, BLOCK, 0, stream>>>(
        x0, y0, mean_f, var_f, gammas, noise, x_tot, y_tot, outp, STEPS);

    dim3 sgrid((STEPS + BLOCK - 1) / BLOCK, NSAMP);
    write_steps<<<sgrid, BLOCK, 0, stream>>>(steps, STEPS);
}